// GCN_En2_27754078666886
// MI455X (gfx1250) — compile-verified
//
#include <hip/hip_runtime.h>

// ---------------------------------------------------------------------------
// GCN 2-layer pipeline for MI455X (gfx1250, wave32).
//   h1 = relu(scatter_add(( x@W1)[src]*w, dst) + b1)
//   out= relu(scatter_add((h1@W2)[src]*w, dst) + b2)
// GEMMs use native f32 WMMA (V_WMMA_F32_16X16X4_F32) keeping exact f32
// numerics; end-to-end the workload is bound by the L2-resident edge
// scatter (f32 atomics), so GEMM precision downgrades buy nothing.
// W1/W2 are pre-packed per call into a WMMA-lane-native layout so the hot
// loop does b64 loads for both A and B (VMEM issue < WMMA issue).
// ---------------------------------------------------------------------------

typedef __attribute__((ext_vector_type(2))) float v2f;
typedef __attribute__((ext_vector_type(8))) float v8f;

#define N_NODES 50000
#define N_EDGES 800000
#define NFEAT 256
#define NHID 128
#define NEMBED 64

// ---------------------------------------------------------------------------
__global__ __launch_bounds__(256) void zero_f32_kernel(float* __restrict__ p, int n) {
    int i = blockIdx.x * blockDim.x + threadIdx.x;
    if (i < n) p[i] = 0.0f;
}

// ---------------------------------------------------------------------------
// Pack B[K x N] (row major) into Bp[K/4][2][N][2]:
//   Bp[(kh*N + c)*2 + j] = B[(kh>>1)*4 + (kh&1)*2 + j][c],  kh = k4*2 + half
// so a WMMA lane (half = lane>>4, c = t*16 + (lane&15)) reads its two
// K-adjacent B values as one coalesced b64.
template<int K, int N>
__global__ __launch_bounds__(256) void pack_B_kernel(
        const float* __restrict__ B, float* __restrict__ Bp) {
    constexpr int nPairs = (K / 2) * N;
    int p = blockIdx.x * blockDim.x + threadIdx.x;
    if (p >= nPairs) return;
    const int c    = p % N;
    const int kh   = p / N;                       // (k4, half) pair id
    const int krow = (kh >> 1) * 4 + (kh & 1) * 2;
    v2f v;
    v.x = B[(size_t)krow * N + c];
    v.y = B[(size_t)(krow + 1) * N + c];
    *(v2f*)(Bp + (size_t)p * 2) = v;
}

// ---------------------------------------------------------------------------
// C[M x (16*NT)] = A[M x K] * Bp (packed), row-major f32, WMMA 16x16x4.
// One wave computes a 16-row strip across the full N width (NT 16-col tiles).
// A (16x4 f32): lane m=lane&15; lanes 0-15 hold K=k,k+1, lanes 16-31 hold
// K=k+2,k+3 -> one b64 per lane per step. B pairs come from Bp as one b64.
// D (16x16 f32, 8 VGPRs): vgpr r -> row r (lanes 0-15) / r+8 (lanes 16-31).
template<int K, int NT>
__global__ __launch_bounds__(256) void gemm_wmma_f32(
        const float* __restrict__ A, const float* __restrict__ Bp,
        float* __restrict__ C, int mTiles) {
    const int lane = threadIdx.x & 31;
    const int tile = blockIdx.x * (blockDim.x >> 5) + (threadIdx.x >> 5);
    if (tile >= mTiles) return;

    constexpr int N = 16 * NT;
    const int m    = lane & 15;   // row within tile (A) / col within tile (B)
    const int half = lane >> 4;   // which K pair of the 4 this lane holds

    const float* Arow = A + (size_t)(tile * 16 + m) * K + half * 2;

    v8f acc[NT];
#pragma unroll
    for (int t = 0; t < NT; ++t) acc[t] = (v8f){0.f,0.f,0.f,0.f,0.f,0.f,0.f,0.f};

    for (int k = 0; k < K; k += 4) {
        v2f a = *(const v2f*)(Arow + k);                     // A[m][k+2h .. +1]
        const float* Bp0 = Bp + ((size_t)((k >> 2) * 2 + half) * N + m) * 2;
#pragma unroll
        for (int t = 0; t < NT; ++t) {
            v2f b = *(const v2f*)(Bp0 + t * 32);             // B[k+2h..+1][t*16+m]
            acc[t] = __builtin_amdgcn_wmma_f32_16x16x4_f32(
                false, a, false, b, (short)0, acc[t], false, false);
        }
    }

    // Store D: vgpr r -> row (tile*16 + r + 8*half), col (t*16 + m)
    float* Cbase = C + (size_t)(tile * 16 + half * 8) * N + m;
#pragma unroll
    for (int t = 0; t < NT; ++t)
#pragma unroll
        for (int r = 0; r < 8; ++r)
            Cbase[(size_t)r * N + t * 16] = acc[t][r];
}

// ---------------------------------------------------------------------------
// One wave per edge; lanes stride the feature dim (F/32 feats per lane).
// Row gathers are explicit b128/b64; scatter-add is global_atomic_add_f32
// into an L2-resident aggregation buffer (192MB L2 >> 25.6MB agg).
template<int F>
__global__ __launch_bounds__(256) void edge_scatter(
        const float* __restrict__ sup, const int* __restrict__ src,
        const int* __restrict__ dst, const float* __restrict__ w,
        float* __restrict__ agg, int nEdges) {
    const int lane = threadIdx.x & 31;
    const int e = blockIdx.x * (blockDim.x >> 5) + (threadIdx.x >> 5);
    if (e >= nEdges) return;

    const int   s  = src[e];
    const int   d  = dst[e];
    const float wt = w[e];

    if constexpr (F == 128) {
        const float4 v = *(const float4*)(sup + (size_t)s * F + lane * 4);
        float* ap = agg + (size_t)d * F + lane * 4;
        atomicAdd(ap + 0, v.x * wt);
        atomicAdd(ap + 1, v.y * wt);
        atomicAdd(ap + 2, v.z * wt);
        atomicAdd(ap + 3, v.w * wt);
    } else {
        const float2 v = *(const float2*)(sup + (size_t)s * F + lane * 2);
        float* ap = agg + (size_t)d * F + lane * 2;
        atomicAdd(ap + 0, v.x * wt);
        atomicAdd(ap + 1, v.y * wt);
    }
}

// ---------------------------------------------------------------------------
__global__ __launch_bounds__(256) void bias_relu_kernel(
        const float* __restrict__ in, const float* __restrict__ bias,
        float* __restrict__ out, int n, int fMask) {
    int i = blockIdx.x * blockDim.x + threadIdx.x;
    if (i < n) {
        float v = in[i] + bias[i & fMask];
        out[i] = v > 0.0f ? v : 0.0f;
    }
}

// ---------------------------------------------------------------------------
extern "C" void kernel_launch(void* const* d_in, const int* in_sizes, int n_in,
                              void* d_out, int out_size, void* d_ws, size_t ws_size,
                              hipStream_t stream) {
    const float* x  = (const float*)d_in[0];
    const int*   ei = (const int*)d_in[1];          // [2, E] flat: src then dst
    const float* ew = (const float*)d_in[2];
    const float* W1 = (const float*)d_in[3];
    const float* b1 = (const float*)d_in[4];
    const float* W2 = (const float*)d_in[5];
    const float* b2 = (const float*)d_in[6];
    float* out = (float*)d_out;                     // [N_NODES, NEMBED]

    const int* src = ei;
    const int* dst = ei + N_EDGES;

    // Workspace layout (f32):
    //   buf0: support1 (N*NHID), reused as support2 (N*NEMBED) after scatter1
    //   buf1: agg1 -> h1 (N*NHID), in-place bias+relu
    //   Bp1 : packed W1 (NFEAT*NHID)
    //   Bp2 : packed W2 (NHID*NEMBED)
    float* sup  = (float*)d_ws;
    float* agg1 = sup  + (size_t)N_NODES * NHID;
    float* Bp1  = agg1 + (size_t)N_NODES * NHID;
    float* Bp2  = Bp1  + (size_t)NFEAT * NHID;

    const int mTiles = N_NODES / 16;                // 3125, exact
    const int wavesPerBlock = 256 / 32;             // 8 waves per block

    const int nAgg1 = N_NODES * NHID;               // 6.4M
    const int nOut  = N_NODES * NEMBED;             // 3.2M

    // Zero aggregation targets (d_out doubles as agg2; re-zeroed every call).
    zero_f32_kernel<<<(nAgg1 + 255) / 256, 256, 0, stream>>>(agg1, nAgg1);
    zero_f32_kernel<<<(nOut + 255) / 256, 256, 0, stream>>>(out, nOut);

    // Pack weights into WMMA-lane-native pair layout (tiny, once per call).
    pack_B_kernel<NFEAT, NHID>
        <<<((NFEAT / 2) * NHID + 255) / 256, 256, 0, stream>>>(W1, Bp1);
    pack_B_kernel<NHID, NEMBED>
        <<<((NHID / 2) * NEMBED + 255) / 256, 256, 0, stream>>>(W2, Bp2);

    // Layer 1: support1 = x @ W1   (K=256, N=128 -> 8 tiles/wave)
    gemm_wmma_f32<NFEAT, NHID / 16>
        <<<(mTiles + wavesPerBlock - 1) / wavesPerBlock, 256, 0, stream>>>(
            x, Bp1, sup, mTiles);

    // agg1 += support1[src] * w  (scatter-add over edges)
    edge_scatter<NHID>
        <<<(N_EDGES + wavesPerBlock - 1) / wavesPerBlock, 256, 0, stream>>>(
            sup, src, dst, ew, agg1, N_EDGES);

    // h1 = relu(agg1 + b1), in place
    bias_relu_kernel<<<(nAgg1 + 255) / 256, 256, 0, stream>>>(
        agg1, b1, agg1, nAgg1, NHID - 1);

    // Layer 2: support2 = h1 @ W2   (K=128, N=64 -> 4 tiles/wave), reuse buf0
    gemm_wmma_f32<NHID, NEMBED / 16>
        <<<(mTiles + wavesPerBlock - 1) / wavesPerBlock, 256, 0, stream>>>(
            agg1, Bp2, sup, mTiles);

    // out += support2[src] * w
    edge_scatter<NEMBED>
        <<<(N_EDGES + wavesPerBlock - 1) / wavesPerBlock, 256, 0, stream>>>(
            sup, src, dst, ew, out, N_EDGES);

    // out = relu(out + b2), in place
    bias_relu_kernel<<<(nOut + 255) / 256, 256, 0, stream>>>(
        out, b2, out, nOut, NEMBED - 1);
}